// DetectPeaksTM_76012331205179
// MI455X (gfx1250) — compile-verified
//
#include <hip/hip_runtime.h>
#include <hip/hip_bf16.h>
#include <stdint.h>

// DetectPeaks: |x|, sliding max (K=301, same pad), keep local maxima, top-2 per row.
// One block per row (6144 rows of 8192 fp32). Bandwidth-bound: 201MB @ 23.3TB/s ~ 8.6us.
// CDNA5 path: double-buffered async global->LDS staging (ASYNCcnt) with one-tile
// prefetch, Gil-Werman sliding max in LDS, key-packed top-2 reduction.

#define NT        8192
#define PAD       150
#define NTILES    4
#define TILE      (NT / NTILES)          // 2048
#define HALO      160                    // >= PAD, multiple of 32
#define BL        (TILE + 2 * HALO)      // 2368 floats staged per tile
#define CHUNK     32
#define NCH       (BL / CHUNK)           // 74 chunks
#define PSTRIDE   36                     // padded chunk stride: 16B-aligned b128 targets
#define PHYS_BL   (NCH * PSTRIDE)        // 2664
#define NTHR      256
#define NGROUPS   (BL / 4)               // 592 b128 groups per tile

__device__ __forceinline__ uint32_t lphys(uint32_t b) {
  // buffer index -> padded LDS index (chunk of 32 floats stored with stride 36)
  return (b >> 5) * PSTRIDE + (b & 31u);
}

// Issue async global->LDS copies for one tile (+halo). In-row 16B groups only;
// out-of-row halo is synthesized as -inf in the chunk pass.
__device__ __forceinline__ void stage_tile_async(float* dst, const float* row,
                                                 int g0, int tid) {
  #pragma unroll
  for (int k = 0; k < 3; ++k) {
    int grp = k * NTHR + tid;                 // 16-byte group id
    if (grp < NGROUPS) {
      int g = g0 + grp * 4;                   // global float index of group start
      if (g >= 0 && g + 4 <= NT) {            // groups never straddle the row edge
        uint32_t lds_off = (uint32_t)(uintptr_t)(&dst[lphys((uint32_t)(grp * 4))]);
        const float* gp = row + g;
        asm volatile("global_load_async_to_lds_b128 %0, %1, off"
                     :: "v"(lds_off), "v"(gp)
                     : "memory");
      }
    }
  }
}

__device__ __forceinline__ void wait_async0() {
#if __has_builtin(__builtin_amdgcn_s_wait_asynccnt)
  __builtin_amdgcn_s_wait_asynccnt(0);
#else
  asm volatile("s_wait_asynccnt 0x0" ::: "memory");
#endif
}

__global__ __launch_bounds__(NTHR)
void detect_peaks_kernel(const float* __restrict__ x,
                         float* __restrict__ out_val,
                         int* __restrict__ out_idx) {
  __shared__ float s_x0[PHYS_BL];
  __shared__ float s_x1[PHYS_BL];
  __shared__ float s_pre[PHYS_BL];
  __shared__ float s_suf[PHYS_BL];
  __shared__ float s_cmax[NCH];
  __shared__ unsigned long long s_k1[NTHR];
  __shared__ unsigned long long s_k2[NTHR];

  const int tid = (int)threadIdx.x;
  const float* row = x + (size_t)blockIdx.x * NT;

  unsigned long long k1 = 0ull, k2 = 0ull;  // top-2 keys, accumulated across tiles

  // Kick off tile 0 before the pipeline loop.
  stage_tile_async(s_x0, row, -HALO, tid);

  for (int t = 0; t < NTILES; ++t) {
    const int g0 = t * TILE - HALO;          // global index of buffer element 0
    wait_async0();                            // tile t resident (this wave's issues)
    __syncthreads();                          // all waves: tile t ready; prev reads done

    float* cur = (t & 1) ? s_x1 : s_x0;
    float* nxt = (t & 1) ? s_x0 : s_x1;
    if (t + 1 < NTILES)                       // prefetch next tile behind compute
      stage_tile_async(nxt, row, (t + 1) * TILE - HALO, tid);

    // ---- per-chunk |x| + halo fill + prefix/suffix/chunk max, fused ----
    if (tid < NCH) {
      uint32_t base = (uint32_t)tid * PSTRIDE;
      int gbase = g0 + tid * CHUNK;
      float m = -INFINITY;
      #pragma unroll
      for (int j = 0; j < CHUNK; ++j) {
        int g = gbase + j;
        float v = (g >= 0 && g < NT) ? fabsf(cur[base + j]) : -INFINITY;
        cur[base + j] = v;                    // cleaned |x| for the scoring pass
        m = fmaxf(m, v);
        s_pre[base + j] = m;
      }
      s_cmax[tid] = m;
      m = -INFINITY;
      #pragma unroll
      for (int j = CHUNK - 1; j >= 0; --j) {
        m = fmaxf(m, cur[base + j]);          // same-thread re-read, no race
        s_suf[base + j] = m;
      }
    }
    __syncthreads();

    // ---- sliding max (Gil-Werman), peak mask, per-thread top-2 ----
    #pragma unroll
    for (int k = 0; k < TILE / NTHR; ++k) {
      int i = t * TILE + k * NTHR + tid;      // global time index (always < NT)
      int l = i - PAD; if (l < 0) l = 0;
      int r = i + PAD; if (r > NT - 1) r = NT - 1;
      int bl = l - g0;
      int br = r - g0;
      int cl = bl >> 5, cr = br >> 5;
      float m = fmaxf(s_suf[lphys((uint32_t)bl)], s_pre[lphys((uint32_t)br)]);
      for (int c = cl + 1; c < cr; ++c)       // ~8-9 broadcast reads
        m = fmaxf(m, s_cmax[c]);
      float xv = cur[lphys((uint32_t)(i - g0))];
      float score = (xv == m) ? xv : 0.0f;    // exact local-max test, zero elsewhere
      // score >= 0 so float bits are order-monotonic; ~i -> lowest index wins ties
      unsigned long long key =
          ((unsigned long long)__float_as_uint(score) << 32) |
          (unsigned long long)(~(uint32_t)i);
      if (key > k1)      { k2 = k1; k1 = key; }
      else if (key > k2) { k2 = key; }
    }
  }

  // ---- block-wide top-2 merge ----
  s_k1[tid] = k1;
  s_k2[tid] = k2;
  __syncthreads();
  for (int s = NTHR / 2; s > 0; s >>= 1) {
    if (tid < s) {
      unsigned long long a1 = s_k1[tid],     a2 = s_k2[tid];
      unsigned long long b1 = s_k1[tid + s], b2 = s_k2[tid + s];
      unsigned long long m1, m2;
      if (a1 >= b1) { m1 = a1; m2 = (b1 > a2) ? b1 : a2; }
      else          { m1 = b1; m2 = (a1 > b2) ? a1 : b2; }
      s_k1[tid] = m1; s_k2[tid] = m2;
    }
    __syncthreads();
  }
  if (tid == 0) {
    unsigned long long r1 = s_k1[0], r2 = s_k2[0];
    out_val[(size_t)blockIdx.x * 2 + 0] = __uint_as_float((uint32_t)(r1 >> 32));
    out_val[(size_t)blockIdx.x * 2 + 1] = __uint_as_float((uint32_t)(r2 >> 32));
    out_idx[(size_t)blockIdx.x * 2 + 0] = (int)(~(uint32_t)r1);
    out_idx[(size_t)blockIdx.x * 2 + 1] = (int)(~(uint32_t)r2);
  }
}

extern "C" void kernel_launch(void* const* d_in, const int* in_sizes, int n_in,
                              void* d_out, int out_size, void* d_ws, size_t ws_size,
                              hipStream_t stream) {
  (void)n_in; (void)d_ws; (void)ws_size; (void)out_size;
  const float* xcorr = (const float*)d_in[0];   // [32,3,64,8192] fp32
  // d_in[1] = nlag, unused by the reference computation.
  const int rows = in_sizes[0] / NT;            // 6144
  // Tuple output concatenated flat: [rows*2] fp32 scores, then [rows*2] int32 indices.
  float* out_val = (float*)d_out;
  int*   out_idx = (int*)((float*)d_out + (size_t)rows * 2);
  detect_peaks_kernel<<<rows, NTHR, 0, stream>>>(xcorr, out_val, out_idx);
}